// GCN_38216618999855
// MI455X (gfx1250) — compile-verified
//
#include <hip/hip_runtime.h>

typedef __attribute__((ext_vector_type(2))) float v2f;
typedef __attribute__((ext_vector_type(8))) float v8f;

#define N_NODES 50000
#define FDIM    128
#define N_EDGES 800000

// ---------------------------------------------------------------------------
// 0) init: deg=1.0 (self loop), agg1=0, agg2=0
// ---------------------------------------------------------------------------
__global__ __launch_bounds__(256) void gcn_init(float* __restrict__ deg,
                                                float* __restrict__ agg1,
                                                float* __restrict__ agg2) {
  long long t = (long long)blockIdx.x * 256 + threadIdx.x;
  if (t < (long long)N_NODES * FDIM) agg1[t] = 0.0f;
  if (t < N_NODES) { deg[t] = 1.0f; agg2[t] = 0.0f; }
}

// ---------------------------------------------------------------------------
// 1) degree accumulation over dst (edge weight 1)
// ---------------------------------------------------------------------------
__global__ __launch_bounds__(256) void gcn_degree(const long long* __restrict__ dst,
                                                  float* __restrict__ deg) {
  int e = blockIdx.x * 256 + threadIdx.x;
  if (e >= N_EDGES) return;
  atomicAdd(&deg[(int)dst[e]], 1.0f);
}

// ---------------------------------------------------------------------------
// 2) dinv = rsqrt(deg)
// ---------------------------------------------------------------------------
__global__ __launch_bounds__(256) void gcn_dinv(const float* __restrict__ deg,
                                                float* __restrict__ dinv) {
  int i = blockIdx.x * 256 + threadIdx.x;
  if (i >= N_NODES) return;
  dinv[i] = rsqrtf(deg[i]);
}

// ---------------------------------------------------------------------------
// 3) h1 = x @ W1 via V_WMMA_F32_16X16X4_F32 (exact fp32 on matrix cores)
//    block = 256 thr = 8 waves; wave w -> 16x16 tile (blockIdx node strip, cols 16w..)
//    A 16x4 layout: lanes 0-15 hold K={k,k+1}, lanes 16-31 K={k+2,k+3}, M=lane&15
//    B 4x16 layout: same K split, N=lane&15
//    C 16x16 layout: VGPR r -> M = r + 8*(lane>>4), N = lane&15
// ---------------------------------------------------------------------------
__global__ __launch_bounds__(256) void gcn_gemm1_wmma(const float* __restrict__ x,
                                                      const float* __restrict__ W1,
                                                      float* __restrict__ h1) {
  const int lane = threadIdx.x & 31;
  const int wave = threadIdx.x >> 5;          // 0..7 -> column tile
  const int row0 = blockIdx.x * 16;           // node strip base (50000/16 = 3125)
  const int col0 = wave * 16;
  const int mn   = lane & 15;                 // M for A, N for B/C
  const int koff = (lane >> 4) << 1;          // 0 or 2

  const float* __restrict__ xrow = x + (size_t)(row0 + mn) * FDIM;

  v8f c = {0.f, 0.f, 0.f, 0.f, 0.f, 0.f, 0.f, 0.f};
#pragma unroll
  for (int k = 0; k < FDIM; k += 4) {
    v2f a, b;
    a.x = xrow[k + koff];
    a.y = xrow[k + koff + 1];
    b.x = W1[(size_t)(k + koff)     * FDIM + col0 + mn];
    b.y = W1[(size_t)(k + koff + 1) * FDIM + col0 + mn];
    c = __builtin_amdgcn_wmma_f32_16x16x4_f32(
        /*neg_a=*/false, a, /*neg_b=*/false, b,
        /*c_mod=*/(short)0, c, /*reuse_a=*/false, /*reuse_b=*/false);
  }

  const int rbase = (lane >> 4) << 3;         // 0 or 8
#pragma unroll
  for (int r = 0; r < 8; ++r) {
    h1[(size_t)(row0 + rbase + r) * FDIM + col0 + mn] = c[r];
  }
}

// ---------------------------------------------------------------------------
// 4) edge scatter, layer 1: one wave per edge, lane handles 4 features (float4)
//    agg1[dst] += h1[src] * dinv[src]*dinv[dst]
// ---------------------------------------------------------------------------
__global__ __launch_bounds__(256) void gcn_scatter1(const long long* __restrict__ src,
                                                    const long long* __restrict__ dst,
                                                    const float* __restrict__ dinv,
                                                    const float* __restrict__ h1,
                                                    float* __restrict__ agg1) {
  const int lane = threadIdx.x & 31;
  const int e = blockIdx.x * 8 + (threadIdx.x >> 5);
  if (e >= N_EDGES) return;
  const int s = (int)src[e];
  const int d = (int)dst[e];
  const float w = dinv[s] * dinv[d];
  const float4 v = *(const float4*)(h1 + (size_t)s * FDIM + lane * 4);
  float* o = agg1 + (size_t)d * FDIM + lane * 4;
  atomicAdd(o + 0, v.x * w);
  atomicAdd(o + 1, v.y * w);
  atomicAdd(o + 2, v.z * w);
  atomicAdd(o + 3, v.w * w);
}

// ---------------------------------------------------------------------------
// 5) layer-1 epilogue: self-loop + bias + ReLU, result in-place into agg1
// ---------------------------------------------------------------------------
__global__ __launch_bounds__(256) void gcn_epilogue1(float* __restrict__ agg1,
                                                     const float* __restrict__ h1,
                                                     const float* __restrict__ dinv,
                                                     const float* __restrict__ b1) {
  long long t = (long long)blockIdx.x * 256 + threadIdx.x;
  if (t >= (long long)N_NODES * FDIM) return;
  const int i = (int)(t >> 7);
  const int f = (int)(t & 127);
  const float di = dinv[i];
  const float v = agg1[t] + h1[t] * (di * di) + b1[f];
  agg1[t] = fmaxf(v, 0.0f);   // agg1 now holds post-ReLU h1
}

// ---------------------------------------------------------------------------
// 6) h2[i] = dot(h1_relu[i,:], W2[:,0]) : one wave per node, shfl_xor reduce
// ---------------------------------------------------------------------------
__global__ __launch_bounds__(256) void gcn_h2(const float* __restrict__ h1r,
                                              const float* __restrict__ W2,
                                              float* __restrict__ h2) {
  const int lane = threadIdx.x & 31;
  const int node = blockIdx.x * 8 + (threadIdx.x >> 5);
  if (node >= N_NODES) return;
  const float4 a = *(const float4*)(h1r + (size_t)node * FDIM + lane * 4);
  const float4 b = *(const float4*)(W2 + lane * 4);
  float p = a.x * b.x + a.y * b.y + a.z * b.z + a.w * b.w;
#pragma unroll
  for (int off = 16; off > 0; off >>= 1) p += __shfl_xor(p, off, 32);
  if (lane == 0) h2[node] = p;
}

// ---------------------------------------------------------------------------
// 7) edge scatter, layer 2 (scalar): agg2[dst] += h2[src]*dinv[src]*dinv[dst]
// ---------------------------------------------------------------------------
__global__ __launch_bounds__(256) void gcn_scatter2(const long long* __restrict__ src,
                                                    const long long* __restrict__ dst,
                                                    const float* __restrict__ dinv,
                                                    const float* __restrict__ h2,
                                                    float* __restrict__ agg2) {
  int e = blockIdx.x * 256 + threadIdx.x;
  if (e >= N_EDGES) return;
  const int s = (int)src[e];
  const int d = (int)dst[e];
  atomicAdd(&agg2[d], h2[s] * dinv[s] * dinv[d]);
}

// ---------------------------------------------------------------------------
// 8) out[i] = agg2[i] + h2[i]*dinv[i]^2 + b2
// ---------------------------------------------------------------------------
__global__ __launch_bounds__(256) void gcn_final(const float* __restrict__ agg2,
                                                 const float* __restrict__ h2,
                                                 const float* __restrict__ dinv,
                                                 const float* __restrict__ b2,
                                                 float* __restrict__ out) {
  int i = blockIdx.x * 256 + threadIdx.x;
  if (i >= N_NODES) return;
  const float di = dinv[i];
  out[i] = agg2[i] + h2[i] * (di * di) + b2[0];
}

// ---------------------------------------------------------------------------
extern "C" void kernel_launch(void* const* d_in, const int* in_sizes, int n_in,
                              void* d_out, int out_size, void* d_ws, size_t ws_size,
                              hipStream_t stream) {
  const float*     x    = (const float*)d_in[0];
  const long long* edge = (const long long*)d_in[1];   // int64 [2, E]
  const float*     W1   = (const float*)d_in[2];
  const float*     b1   = (const float*)d_in[3];
  const float*     W2   = (const float*)d_in[4];
  const float*     b2   = (const float*)d_in[5];
  float*           out  = (float*)d_out;

  const long long* src = edge;
  const long long* dst = edge + N_EDGES;

  // workspace layout (fp32): h1 | agg1 | deg | dinv | h2 | agg2   (~52 MB)
  float* ws   = (float*)d_ws;
  float* h1   = ws;
  float* agg1 = h1   + (size_t)N_NODES * FDIM;
  float* deg  = agg1 + (size_t)N_NODES * FDIM;
  float* dinv = deg  + N_NODES;
  float* h2   = dinv + N_NODES;
  float* agg2 = h2   + N_NODES;

  const int nf_blocks   = (N_NODES * FDIM + 255) / 256;  // 25000
  const int edge_blocks = (N_EDGES + 255) / 256;         // 3125
  const int node_blocks = (N_NODES + 255) / 256;         // 196

  gcn_init     <<<nf_blocks, 256, 0, stream>>>(deg, agg1, agg2);
  gcn_degree   <<<edge_blocks, 256, 0, stream>>>(dst, deg);
  gcn_dinv     <<<node_blocks, 256, 0, stream>>>(deg, dinv);
  gcn_gemm1_wmma<<<N_NODES / 16, 256, 0, stream>>>(x, W1, h1);   // 3125 blocks
  gcn_scatter1 <<<N_EDGES / 8, 256, 0, stream>>>(src, dst, dinv, h1, agg1);
  gcn_epilogue1<<<nf_blocks, 256, 0, stream>>>(agg1, h1, dinv, b1);
  gcn_h2       <<<(N_NODES + 7) / 8, 256, 0, stream>>>(agg1, W2, h2);
  gcn_scatter2 <<<edge_blocks, 256, 0, stream>>>(src, dst, dinv, h2, agg2);
  gcn_final    <<<node_blocks, 256, 0, stream>>>(agg2, h2, dinv, b2, out);
}